// HopfieldNet_6244882448514
// MI455X (gfx1250) — compile-verified
//
#include <hip/hip_runtime.h>
#include <hip/hip_bf16.h>

typedef __attribute__((ext_vector_type(2))) float v2f;
typedef __attribute__((ext_vector_type(8))) float v8f;

#define NN    8192
#define TK    256          // K-tile per block pass
#define PITCH 260          // LDS row pitch (dwords), conflict-free for WMMA A reads
#define EPS   1e-6f

// ---------------------------------------------------------------- init: x <- x0, done <- 0
__global__ void k_init(float* __restrict__ x, const float* __restrict__ x_in, int* done) {
    int i = blockIdx.x * blockDim.x + threadIdx.x;
    x[i] = x_in[i];
    if (i == 0) *done = 0;
}

// ---------------------------------------------------------------- sign update: x <- sign(y + b)
__global__ void k_sign(float* __restrict__ x, const float* __restrict__ y,
                       const float* __restrict__ bias, const int* done) {
    if (*done) return;
    int i = blockIdx.x * blockDim.x + threadIdx.x;
    float v = y[i] + bias[i];                 // THRESHOLD == 0
    x[i] = (float)((v > 0.0f) - (v < 0.0f)); // jnp.sign semantics (0 -> 0)
}

// ---------------------------------------------------------------- y = W x via V_WMMA_F32_16X16X4_F32
// Block: 256 threads (8 waves), 16 rows of W. Also emits per-block partials of
// s1 = sum x_r*y_r and s2 = sum b_r*x_r for the deterministic energy reduction.
__global__ void k_matvec(const float* __restrict__ W, const float* __restrict__ x,
                         const float* __restrict__ bias, float* __restrict__ y,
                         float* __restrict__ p1, float* __restrict__ p2, const int* done) {
    if (*done) return;

    __shared__ float tile[16 * PITCH];   // 16 rows x 256 cols (padded)
    __shared__ float wout[8 * 16];       // per-wave row partials

    const int tid  = threadIdx.x;
    const int r0   = blockIdx.x * 16;
    const int lrow = tid >> 4;           // 0..15 : row loaded by this thread
    const int lj   = tid & 15;           // 0..15 : position within row-loader group
    const int wave = tid >> 5;           // 0..7
    const int lid  = tid & 31;
    const int mrow = lid & 15;           // A-fragment M row
    const int koff = (lid >> 4) << 1;    // 0 (lanes 0-15 hold K=0,1) or 2 (lanes 16-31 hold K=2,3)

    const float* wrow = W + (size_t)(r0 + lrow) * NN;

    v8f acc = {};                        // 16x16 f32 C/D fragment

    for (int k0 = 0; k0 < NN; k0 += TK) {
        // ---- cooperative coalesced tile load: 16 threads per row, 4 float4 each
#pragma unroll
        for (int q = 0; q < 4; ++q) {
            int col = (lj + q * 16) * 4;
            float4 v = *(const float4*)(wrow + k0 + col);
            *(float4*)(&tile[lrow * PITCH + col]) = v;
        }
        // prefetch next tile while we compute on this one (global_prefetch_b8)
        if (k0 + TK < NN) {
#pragma unroll
            for (int q = 0; q < 4; ++q)
                __builtin_prefetch(wrow + k0 + TK + (lj + q * 16) * 4, 0, 1);
        }
        __syncthreads();

        // ---- each wave: 8 WMMAs over its 32-wide K strip
        const int kbase = wave * 32;
#pragma unroll
        for (int s = 0; s < 8; ++s) {
            int kk = kbase + s * 4;
            v2f a, b;
            // A = W tile, 16x4, ISA lane layout (K packed 0/1 low half, 2/3 high half)
            a.x = tile[mrow * PITCH + kk + koff];
            a.y = tile[mrow * PITCH + kk + koff + 1];
            // B = x chunk broadcast to all 16 columns: B[k][n] = x[k0+kk+k]
            b.x = x[k0 + kk + koff];
            b.y = x[k0 + kk + koff + 1];
            acc = __builtin_amdgcn_wmma_f32_16x16x4_f32(
                false, a, false, b, (short)0, acc, false, false);
        }
        __syncthreads();
    }

    // ---- column 0 of C: lane 0 holds rows 0..7 (VGPR v -> M=v), lane 16 rows 8..15
    if (lid == 0 || lid == 16) {
        int base = (lid >> 4) * 8;
#pragma unroll
        for (int v = 0; v < 8; ++v) wout[wave * 16 + base + v] = acc[v];
    }
    __syncthreads();

    // ---- fixed-order cross-wave reduction, emit y and energy partials
    if (tid < 16) {
        float yv = 0.0f;
#pragma unroll
        for (int w = 0; w < 8; ++w) yv += wout[w * 16 + tid];
        int r = r0 + tid;
        y[r] = yv;
        float xr = x[r];
        tile[tid]      = xr * yv;       // reuse LDS as scratch
        tile[16 + tid] = bias[r] * xr;
    }
    __syncthreads();
    if (tid == 0) {
        float s1 = 0.0f, s2 = 0.0f;
#pragma unroll
        for (int i = 0; i < 16; ++i) { s1 += tile[i]; s2 += tile[16 + i]; }
        p1[blockIdx.x] = s1;
        p2[blockIdx.x] = s2;
    }
}

// ---------------------------------------------------------------- deterministic energy reduce + convergence
__global__ void k_energy(const float* __restrict__ p1, const float* __restrict__ p2,
                         float* __restrict__ e_prev, int* __restrict__ done, int init_mode) {
    if (!init_mode && *done) return;
    __shared__ float s1[256], s2[256];
    int t = threadIdx.x;
    s1[t] = p1[t] + p1[t + 256];
    s2[t] = p2[t] + p2[t + 256];
    __syncthreads();
    for (int off = 128; off > 0; off >>= 1) {
        if (t < off) { s1[t] += s1[t + off]; s2[t] += s2[t + off]; }
        __syncthreads();
    }
    if (t == 0) {
        float e_new = -0.5f * s1[0] - s2[0];
        if (!init_mode) {
            if (fabsf(e_new - *e_prev) < EPS) *done = 1;
        }
        *e_prev = e_new;
    }
}

// ----------------------------------------------------------------
extern "C" void kernel_launch(void* const* d_in, const int* in_sizes, int n_in,
                              void* d_out, int out_size, void* d_ws, size_t ws_size,
                              hipStream_t stream) {
    const float* x_in = (const float*)d_in[0];
    const float* W    = (const float*)d_in[1];
    const float* bias = (const float*)d_in[2];

    float* x = (float*)d_out;            // state lives in d_out; final value is the answer

    float* wsf = (float*)d_ws;
    float* y   = wsf;                    // 8192
    float* p1  = wsf + 8192;             // 512 per-block partials (x.y)
    float* p2  = wsf + 8192 + 512;       // 512 per-block partials (b.x)
    float* ep  = wsf + 8192 + 1024;      // scalar e_prev
    int*   dn  = (int*)(wsf + 8192 + 1025);

    const int MATVEC_BLOCKS = NN / 16;   // 512

    // prologue: x <- x0 ; y0 = W x0 ; e0 = -0.5 x0.y0 - b.x0
    k_init  <<<NN / 256, 256, 0, stream>>>(x, x_in, dn);
    k_matvec<<<MATVEC_BLOCKS, 256, 0, stream>>>(W, x, bias, y, p1, p2, dn);
    k_energy<<<1, 256, 0, stream>>>(p1, p2, ep, dn, 1);

    // masked fixed-iteration loop (matches the reference lax.scan exactly)
    for (int it = 0; it < 128; ++it) {
        k_sign  <<<NN / 256, 256, 0, stream>>>(x, y, bias, dn);
        k_matvec<<<MATVEC_BLOCKS, 256, 0, stream>>>(W, x, bias, y, p1, p2, dn);
        k_energy<<<1, 256, 0, stream>>>(p1, p2, ep, dn, 0);
    }
}